// TensorProductLayer_6734508720709
// MI455X (gfx1250) — compile-verified
//
#include <hip/hip_runtime.h>

#define NT_STORE(p, v)  __builtin_nontemporal_store((v), (p))
#define NT_LOAD(p)      __builtin_nontemporal_load(p)

// ---- device-pass feature gates (host pass always takes the plain body so the
// ---- kernel symbol + launch stay consistent across the two compile passes) ----
#if defined(__HIP_DEVICE_COMPILE__) && __has_builtin(__builtin_amdgcn_tensor_load_to_lds) && \
    __has_builtin(__builtin_amdgcn_s_wait_tensorcnt)
#define DEV_TDM 1
#else
#define DEV_TDM 0
#endif
#if DEV_TDM && __has_builtin(__builtin_amdgcn_tensor_store_from_lds)
#define DEV_TDM_STORE 1
#else
#define DEV_TDM_STORE 0
#endif
#if DEV_TDM && __has_builtin(__builtin_amdgcn_wmma_f32_16x16x4_f32)
#define DEV_WMMA 1
#else
#define DEV_WMMA 0
#endif

// ---------------- shared math helpers ----------------
// mm[d,e] = sum_f a2[d,f]*b2[f,e]  (VALU fallback for the WMMA term)
__device__ __forceinline__ void mm33(const float a2[9], const float b2[9], float mm[9]) {
#pragma unroll
    for (int d = 0; d < 3; ++d)
#pragma unroll
        for (int e = 0; e < 3; ++e) {
            float t = a2[d * 3] * b2[e];
            t = fmaf(a2[d * 3 + 1], b2[3 + e], t);
            t = fmaf(a2[d * 3 + 2], b2[6 + e], t);
            mm[d * 3 + e] = t;
        }
}

// out0      = x0*y0 + dot(x1,y1) + frob(x2,y2)
// out1[d]   = x0*y1[d] + y0*x1[d] + sum_e x1[e]*y2[e,d] + sum_e x2[d,e]*y1[e]
// out2[d,e] = mm[d,e] + x0*y2[d,e] + y0*x2[d,e] + x1[d]*y1[e]
__device__ __forceinline__ void tp_math_mm(float a0, float b0,
                                           const float a1[3], const float b1[3],
                                           const float a2[9], const float b2[9],
                                           const float mm[9],
                                           float& r0, float r1[3], float r2[9]) {
    r0 = a0 * b0;
#pragma unroll
    for (int k = 0; k < 3; ++k) r0 = fmaf(a1[k], b1[k], r0);
#pragma unroll
    for (int k = 0; k < 9; ++k) r0 = fmaf(a2[k], b2[k], r0);

#pragma unroll
    for (int d = 0; d < 3; ++d) {
        float t = a0 * b1[d];
        t = fmaf(b0, a1[d], t);
#pragma unroll
        for (int e = 0; e < 3; ++e) {
            t = fmaf(a1[e], b2[e * 3 + d], t);   // x1 . y2
            t = fmaf(a2[d * 3 + e], b1[e], t);   // x2 . y1
        }
        r1[d] = t;
    }
#pragma unroll
    for (int d = 0; d < 3; ++d)
#pragma unroll
        for (int e = 0; e < 3; ++e) {
            float t = mm[d * 3 + e];
            t = fmaf(a0, b2[d * 3 + e], t);
            t = fmaf(b0, a2[d * 3 + e], t);
            t = fmaf(a1[d], b1[e], t);
            r2[d * 3 + e] = t;
        }
}

#if DEV_TDM
// ---------------- TDM descriptor helpers (CDNA5 ISA ch.8) ----------------
typedef unsigned int v4u __attribute__((ext_vector_type(4)));
typedef int          v8i __attribute__((ext_vector_type(8)));
typedef int          v4i __attribute__((ext_vector_type(4)));
#if DEV_WMMA
typedef float        v2f __attribute__((ext_vector_type(2)));
typedef float        v8f __attribute__((ext_vector_type(8)));
#endif

__device__ __forceinline__ v4u tdm_g0(unsigned lds_byte_off, const void* gptr) {
    unsigned long long ga = (unsigned long long)(size_t)gptr;
    v4u g0;
    g0.x = 1u;                                    // [1:0] count=1 (valid, user mode)
    g0.y = lds_byte_off;                          // [63:32] lds_addr (bytes)
    g0.z = (unsigned)(ga & 0xFFFFFFFFull);        // [95:64] global_addr lo
    g0.w = (unsigned)((ga >> 32) & 0x01FFFFFFull) // [120:96] global_addr hi
         | 0x80000000u;                           // [127:126] type=2 ("image")
    return g0;
}
__device__ __forceinline__ v8i tdm_g1(unsigned nelem) {
    v8i g1;
    g1[0] = (int)(2u << 16);                      // [17:16] data_size=2 -> 4B elems
    g1[1] = (int)((nelem & 0xFFFFu) << 16);       // [63:48]  tensor_dim0[15:0]
    g1[2] = (int)(((nelem >> 16) & 0xFFFFu)       // [79:64]  tensor_dim0[31:16]
         | (1u << 16));                           // [95:80]  tensor_dim1 = 1
    g1[3] = (int)((nelem & 0xFFFFu) << 16);       // [127:112] tile_dim0 = nelem
    g1[4] = 0;                                    // tile_dim1/2 = 0 (unused)
    g1[5] = (int)nelem;                           // [207:160] tensor_dim0_stride
    g1[6] = 0;
    g1[7] = 0;
    return g1;
}
__device__ __forceinline__ void tdm_load_1d(unsigned lds, const void* g, unsigned n) {
    v4i z4 = {0, 0, 0, 0};
#if defined(__clang_major__) && (__clang_major__ >= 23)
    v8i z8 = {0, 0, 0, 0, 0, 0, 0, 0};
    __builtin_amdgcn_tensor_load_to_lds(tdm_g0(lds, g), tdm_g1(n), z4, z4, z8, 0);
#else
    __builtin_amdgcn_tensor_load_to_lds(tdm_g0(lds, g), tdm_g1(n), z4, z4, 0);
#endif
}
#if DEV_TDM_STORE
__device__ __forceinline__ void tdm_store_1d(unsigned lds, void* g, unsigned n) {
    v4i z4 = {0, 0, 0, 0};
#if defined(__clang_major__) && (__clang_major__ >= 23)
    v8i z8 = {0, 0, 0, 0, 0, 0, 0, 0};
    __builtin_amdgcn_tensor_store_from_lds(tdm_g0(lds, g), tdm_g1(n), z4, z4, z8, 0);
#else
    __builtin_amdgcn_tensor_store_from_lds(tdm_g0(lds, g), tdm_g1(n), z4, z4, 0);
#endif
}
#endif
#endif // DEV_TDM

#define TILE 256
// In-LDS layout (floats): x0[256] y0[256] x1[768] y1[768] x2[2304] y2[2304]
#define IX0 0
#define IY0 256
#define IX1 512
#define IY1 1280
#define IX2 2048
#define IY2 4352
#define INBUF  6656
// Out-LDS layout (floats): o0[256] o1[768] o2[2304]
#define OO0 0
#define OO1 256
#define OO2 1024
#define OUTBUF 3328

// ============================ single kernel symbol ============================
__global__ __launch_bounds__(256) void tp_layer_kernel(
    const float* __restrict__ x0, const float* __restrict__ y0,
    const float* __restrict__ x1, const float* __restrict__ y1,
    const float* __restrict__ x2, const float* __restrict__ y2,
    float* __restrict__ o0, float* __restrict__ o1, float* __restrict__ o2,
    int total)
{
#if DEV_TDM
    __shared__ float sIn[2][INBUF];
    __shared__ float sOut[2][OUTBUF];   // TDM-store staging; o2 region doubles as
                                        // the WMMA mm scratch
    const unsigned lin = (unsigned)(size_t)(void*)&sIn[0][0];
#if DEV_TDM_STORE
    const unsigned lout = (unsigned)(size_t)(void*)&sOut[0][0];
#endif
    const int ntiles  = (total + TILE - 1) / TILE;
    const int gstride = (int)gridDim.x;
    const int j       = (int)threadIdx.x;

    // TDM ops are wave-level and IGNORE EXEC: gate with a scalar branch.
    const bool issuer =
        (__builtin_amdgcn_readfirstlane((int)(threadIdx.x >> 5)) == 0);

    auto issue_loads = [&](int t, int b) {
        const unsigned base = lin + (unsigned)b * (INBUF * 4u);
        const int      rem  = total - t * TILE;
        const unsigned cnt  = (unsigned)(rem < TILE ? rem : TILE);
        const size_t   e    = (size_t)t * TILE;
        tdm_load_1d(base + IX0 * 4u, x0 + e,     cnt);
        tdm_load_1d(base + IY0 * 4u, y0 + e,     cnt);
        tdm_load_1d(base + IX1 * 4u, x1 + e * 3, cnt * 3u);
        tdm_load_1d(base + IY1 * 4u, y1 + e * 3, cnt * 3u);
        tdm_load_1d(base + IX2 * 4u, x2 + e * 9, cnt * 9u);
        tdm_load_1d(base + IY2 * 4u, y2 + e * 9, cnt * 9u);
    };
#if DEV_TDM_STORE
    auto issue_stores = [&](int t, int b) {
        const unsigned base = lout + (unsigned)b * (OUTBUF * 4u);
        const int      rem  = total - t * TILE;
        const unsigned cnt  = (unsigned)(rem < TILE ? rem : TILE);
        const size_t   e    = (size_t)t * TILE;
        tdm_store_1d(base + OO0 * 4u, o0 + e,     cnt);
        tdm_store_1d(base + OO1 * 4u, o1 + e * 3, cnt * 3u);
        tdm_store_1d(base + OO2 * 4u, o2 + e * 9, cnt * 9u);
    };
#endif

    const int t0 = (int)blockIdx.x;
    int buf = 0;
    if (issuer && t0 < ntiles) issue_loads(t0, 0);

    for (int t = t0; t < ntiles; t += gstride) {
        if (issuer) {
#if DEV_TDM_STORE
            // In-order TENSORcnt, issue order L(t),S(t-1),L(t+1),S(t),...:
            // waiting to <=3 retires all loads for tile t while S(t-1) flies.
            if (t == t0) __builtin_amdgcn_s_wait_tensorcnt(0);
            else         __builtin_amdgcn_s_wait_tensorcnt(3);
#else
            __builtin_amdgcn_s_wait_tensorcnt(0);
#endif
        }
        __syncthreads();
        const int tn = t + gstride;
        if (issuer && tn < ntiles) issue_loads(tn, buf ^ 1);

        const int rem = total - t * TILE;
        const int cnt = rem < TILE ? rem : TILE;
        const float* S = &sIn[buf][0];
        float*       T = &sOut[buf][0];

#if DEV_WMMA
        // ---- v_wmma_f32_16x16x4_f32: block-diagonal batched 3x3 matmuls ----
        // 5 elements per WMMA at rows/cols 3p..3p+2; 7 WMMAs cover this wave's
        // 32 elements. Runs under full EXEC (no divergent guard around it).
        {
            const int  lane  = j & 31;
            const int  m     = lane & 15;       // A-row / B-col owned by lane
            const int  half  = lane >> 4;       // K-half striping
            const int  p     = m / 3;           // 3x3 block id
            const int  dm    = m - 3 * p;       // d for A-row, e for B-col
            const int  wbase = (j >> 5) * 32;   // wave's element base in tile
            const bool vm    = (m < 15);
#pragma unroll
            for (int q = 0; q < 7; ++q) {
                const int  lp = q * 5 + p;
                const bool pv = vm && (lp < 32);
                const int  el = wbase + lp;
                const int  ab = IX2 + el * 9 + dm * 3;  // a2 row d, k=0..2
                const int  bb = IY2 + el * 9 + dm;      // b2 col e, k stride 3
                v2f A, B;
                A.x = pv ? S[ab + 2 * half] : 0.0f;           // A[m, 2h]
                A.y = (pv && half == 0) ? S[ab + 1] : 0.0f;   // A[m, 1]; K=3 -> 0
                B.x = pv ? S[bb + 6 * half] : 0.0f;           // B[2h, m]
                B.y = (pv && half == 0) ? S[bb + 3] : 0.0f;   // B[1, m]; K=3 -> 0
                v8f C = {};
                v8f D = __builtin_amdgcn_wmma_f32_16x16x4_f32(
                            false, A, false, B, (short)0, C, false, false);
                // Scatter in-block diagonal entries D[3p+d, 3p+e=m] to LDS.
                // Readers are lanes of the SAME wave; LDS is in-order per wave.
#pragma unroll
                for (int r = 0; r < 8; ++r) {
                    const int M = r + 8 * half;   // D-row held by this lane
                    if (pv && M >= 3 * p && M < 3 * p + 3 && M < 15)
                        T[OO2 + el * 9 + (M - 3 * p) * 3 + dm] = D[r];
                }
            }
        }
#endif
        if (j < cnt) {
            const float a0 = S[IX0 + j];
            const float b0 = S[IY0 + j];
            float a1[3], b1[3], a2[9], b2[9];
#pragma unroll
            for (int k = 0; k < 3; ++k) { a1[k] = S[IX1 + j * 3 + k]; b1[k] = S[IY1 + j * 3 + k]; }
#pragma unroll
            for (int k = 0; k < 9; ++k) { a2[k] = S[IX2 + j * 9 + k]; b2[k] = S[IY2 + j * 9 + k]; }

            float mm[9];
#if DEV_WMMA
#pragma unroll
            for (int k = 0; k < 9; ++k) mm[k] = T[OO2 + j * 9 + k];  // same-wave writes
#else
            mm33(a2, b2, mm);
#endif
            float r0, r1[3], r2[9];
            tp_math_mm(a0, b0, a1, b1, a2, b2, mm, r0, r1, r2);

#if DEV_TDM_STORE
            T[OO0 + j] = r0;
#pragma unroll
            for (int k = 0; k < 3; ++k) T[OO1 + j * 3 + k] = r1[k];
#pragma unroll
            for (int k = 0; k < 9; ++k) T[OO2 + j * 9 + k] = r2[k];
#else
            const size_t i  = (size_t)t * TILE + j;
            NT_STORE(o0 + i, r0);
#pragma unroll
            for (int k = 0; k < 3; ++k) NT_STORE(o1 + i * 3 + k, r1[k]);
#pragma unroll
            for (int k = 0; k < 9; ++k) NT_STORE(o2 + i * 9 + k, r2[k]);
#endif
        }
#if DEV_TDM_STORE
        __syncthreads();                   // out-tile complete across all waves
        if (issuer) issue_stores(t, buf);  // async LDS->HBM; drained by the <=3
                                           // wait two iters later / s_endpgm wait-idle
#endif
        buf ^= 1;
    }
#else
    // ---------------- plain body (host pass / no-TDM toolchains) ----------------
    const int stride = (int)(gridDim.x * blockDim.x);
    for (int i = (int)(blockIdx.x * blockDim.x + threadIdx.x); i < total; i += stride) {
        const size_t i3 = (size_t)i * 3, i9 = (size_t)i * 9;
        const float a0 = NT_LOAD(x0 + i);
        const float b0 = NT_LOAD(y0 + i);
        float a1[3], b1[3], a2[9], b2[9];
#pragma unroll
        for (int k = 0; k < 3; ++k) { a1[k] = NT_LOAD(x1 + i3 + k); b1[k] = NT_LOAD(y1 + i3 + k); }
#pragma unroll
        for (int k = 0; k < 9; ++k) { a2[k] = NT_LOAD(x2 + i9 + k); b2[k] = NT_LOAD(y2 + i9 + k); }
        float mm[9];
        mm33(a2, b2, mm);
        float r0, r1[3], r2[9];
        tp_math_mm(a0, b0, a1, b1, a2, b2, mm, r0, r1, r2);
        NT_STORE(o0 + i, r0);
#pragma unroll
        for (int k = 0; k < 3; ++k) NT_STORE(o1 + i3 + k, r1[k]);
#pragma unroll
        for (int k = 0; k < 9; ++k) NT_STORE(o2 + i9 + k, r2[k]);
    }
#endif
}

extern "C" void kernel_launch(void* const* d_in, const int* in_sizes, int n_in,
                              void* d_out, int out_size, void* d_ws, size_t ws_size,
                              hipStream_t stream) {
    // setup_inputs() insertion order: x0, y0, x1, y1, x2, y2
    const float* x0 = (const float*)d_in[0];
    const float* y0 = (const float*)d_in[1];
    const float* x1 = (const float*)d_in[2];
    const float* y1 = (const float*)d_in[3];
    const float* x2 = (const float*)d_in[4];
    const float* y2 = (const float*)d_in[5];

    const int total = in_sizes[0];           // N*C
    float* o0 = (float*)d_out;               // N*C
    float* o1 = o0 + (size_t)total;          // N*C*3
    float* o2 = o0 + (size_t)total * 4;      // N*C*9

    const int ntiles = (total + TILE - 1) / TILE;
    int blocks = ntiles < 8192 ? ntiles : 8192;
    tp_layer_kernel<<<blocks, TILE, 0, stream>>>(x0, y0, x1, y1, x2, y2,
                                                 o0, o1, o2, total);
}